// VoxelBackBone8xFusion_66726611910699
// MI455X (gfx1250) — compile-verified
//
#include <hip/hip_runtime.h>
#include <hip/hip_bf16.h>

typedef __attribute__((ext_vector_type(2))) float v2f;
typedef __attribute__((ext_vector_type(8))) float v8f;

__device__ __forceinline__ int iclamp(int x, int lo, int hi) {
    return x < lo ? lo : (x > hi ? hi : x);
}

// Repack weights (K, CIN, COUT) -> (K, CIN/2, COUT, 2) so the B-tile fetch in the
// conv kernel (two consecutive Cin rows per lane) becomes one aligned b64 load.
__global__ __launch_bounds__(256) void repack_w(
    const float* __restrict__ src, float* __restrict__ dst,
    int K, int CIN, int COUT) {
    int s = blockIdx.x * blockDim.x + threadIdx.x;
    int total = K * CIN * COUT;
    if (s >= total) return;
    int n = s % COUT;
    int t = s / COUT;
    int cin = t % CIN;
    int k = t / CIN;
    dst[(((size_t)k * (CIN >> 1) + (cin >> 1)) * COUT + n) * 2 + (cin & 1)] = src[s];
}

// One wave (32 lanes) computes a full 16-row x COUT output tile:
//   rows [row0, row0+16), ALL output channels (CTILES = COUT/16 accumulator sets).
// Each gathered A chunk (16 rows x 4 cin, one b64 load) is reused across CTILES
// WMMAs; each B chunk is one b64 load from the pair-packed weights. Accumulation
// via V_WMMA_F32_16X16X4_F32 (exact fp32).
template <int CIN, int COUT, int K>
__global__ __launch_bounds__(256) void spconv_wmma(
    const float* __restrict__ feats,  // (Nin, CIN)
    const int* __restrict__ tbl,      // (Nout, K), -1 = no neighbor
    const float* __restrict__ Wpk,    // (K, CIN/2, COUT, 2) pair-packed
    const float* __restrict__ gamma,  // (COUT)
    const float* __restrict__ beta,   // (COUT)
    float* __restrict__ out,          // (Nout, COUT)
    int Nout) {
    constexpr int CTILES = COUT / 16;
    const int lane = threadIdx.x & 31;
    const int wave = threadIdx.x >> 5;
    const int wave_global = blockIdx.x * (blockDim.x >> 5) + wave;
    const int row_tiles = (Nout + 15) >> 4;
    if (wave_global >= row_tiles) return;  // whole-wave exit: EXEC stays full for WMMA

    const int row0 = wave_global * 16;
    const int m = lane & 15;  // output row within tile (A) / output col within ctile (B, C/D)
    const int h = lane >> 4;  // half-wave selector
    const int r = row0 + m;
    const bool rvalid = (r < Nout);

    v8f acc[CTILES];
#pragma unroll
    for (int ct = 0; ct < CTILES; ++ct) acc[ct] = (v8f){};

#pragma unroll 1
    for (int k = 0; k < K; ++k) {
        int idx = rvalid ? tbl[(size_t)r * K + k] : -1;
        // uniform skip when this kernel offset has no neighbors anywhere in the tile
        if (__ballot(idx >= 0) == 0ull) continue;

        const float* fp = (idx >= 0) ? (feats + (size_t)idx * CIN) : nullptr;
        // per-lane B base for this k: pair row h, column m (ct/cin offsets added below)
        const float* wk = Wpk + (((size_t)k * (CIN / 2) + h) * COUT + m) * 2;

#pragma unroll
        for (int cin0 = 0; cin0 < CIN; cin0 += 4) {
            // A tile 16x4: lane holds row m, K-elems {2h, 2h+1} -> one 8B load
            v2f a;
            if (idx >= 0) {
                a = *(const v2f*)(fp + cin0 + 2 * h);
            } else {
                a.x = 0.0f;
                a.y = 0.0f;
            }
#pragma unroll
            for (int ct = 0; ct < CTILES; ++ct) {
                // B tile 4x16: pair (cin0/2 + h), col (ct*16 + m) -> one 8B load
                v2f b = *(const v2f*)(wk + ((size_t)(cin0 >> 1) * COUT + ct * 16) * 2);
                acc[ct] = __builtin_amdgcn_wmma_f32_16x16x4_f32(
                    false, a, false, b, (short)0, acc[ct], false, false);
            }
        }
    }

    // BN + ReLU epilogue; C/D layout: VGPR j holds M = j + 8h, N = m
#pragma unroll
    for (int ct = 0; ct < CTILES; ++ct) {
        const float g = gamma[ct * 16 + m];
        const float bi = beta[ct * 16 + m];
#pragma unroll
        for (int j = 0; j < 8; ++j) {
            int rr = row0 + j + 8 * h;
            if (rr < Nout) {
                float v = acc[ct][j] * g + bi;
                out[(size_t)rr * COUT + ct * 16 + m] = v > 0.0f ? v : 0.0f;
            }
        }
    }
}

// x1[i, c] += valid(i) ? bilinear_upsample4x(img)[c, v_i, u_i] : 0
// img is (16, 96, 320); jax.image.resize bilinear upscale uses half-pixel centers
// with edge clamp: src = (dst + 0.5)/4 - 0.5.
__global__ __launch_bounds__(256) void img_fuse(
    float* __restrict__ x1, const float* __restrict__ img,
    const int* __restrict__ pix, int N) {
    const int IH = 96, IW = 320;
    int i = blockIdx.x * blockDim.x + threadIdx.x;
    int vox = i >> 4;
    int c = i & 15;
    if (vox >= N) return;
    if (pix[vox * 4 + 3] <= 0) return;
    int vy = pix[vox * 4 + 1];
    int ux = pix[vox * 4 + 2];
    float fy = (vy + 0.5f) * 0.25f - 0.5f;
    float fx = (ux + 0.5f) * 0.25f - 0.5f;
    float yf = floorf(fy), xf = floorf(fx);
    float wy = fy - yf, wx = fx - xf;
    int y0 = iclamp((int)yf, 0, IH - 1), y1 = iclamp((int)yf + 1, 0, IH - 1);
    int x0 = iclamp((int)xf, 0, IW - 1), x1c = iclamp((int)xf + 1, 0, IW - 1);
    const float* p = img + (size_t)c * IH * IW;
    float v00 = p[y0 * IW + x0], v01 = p[y0 * IW + x1c];
    float v10 = p[y1 * IW + x0], v11 = p[y1 * IW + x1c];
    float val = v00 * (1.0f - wy) * (1.0f - wx) + v01 * (1.0f - wy) * wx +
                v10 * wy * (1.0f - wx) + v11 * wy * wx;
    x1[(size_t)vox * 16 + c] += val;
}

#define LAUNCH_SPCONV(CIN, COUT, K, FEATS, TBL, LW, LG, LB, OUT, NOUT)          \
    do {                                                                        \
        int tiles_ = ((NOUT) + 15) / 16;                                        \
        int blocks_ = (tiles_ + 7) / 8;                                         \
        spconv_wmma<CIN, COUT, K><<<blocks_, 256, 0, stream>>>(                 \
            FEATS, TBL, LW, LG, LB, OUT, NOUT);                                 \
    } while (0)

extern "C" void kernel_launch(void* const* d_in, const int* in_sizes, int n_in,
                              void* d_out, int out_size, void* d_ws, size_t ws_size,
                              hipStream_t stream) {
    const float* voxel_feats = (const float*)d_in[0];
    const float* img = (const float*)d_in[1];

    // params: 36 arrays (12 layers x {w,g,b}); canonical layer order:
    // in, c1, c2a, c2b, c2c, c3a, c3b, c3c, c4a, c4b, c4c, out
    const float *Wp[12], *Gp[12], *Bp[12];
    if (in_sizes[2] == 27 * 4 * 16) {
        // dict insertion order: w_l, g_l, b_l per layer
        for (int i = 0; i < 12; ++i) {
            Wp[i] = (const float*)d_in[2 + 3 * i];
            Gp[i] = (const float*)d_in[3 + 3 * i];
            Bp[i] = (const float*)d_in[4 + 3 * i];
        }
    } else {
        // jax pytree sorted-key order: all b_* then g_* then w_*,
        // each in sorted layer order: c1,c2a,c2b,c2c,c3a,c3b,c3c,c4a,c4b,c4c,in,out
        const int perm[12] = {10, 0, 1, 2, 3, 4, 5, 6, 7, 8, 9, 11};
        for (int i = 0; i < 12; ++i) {
            Bp[i] = (const float*)d_in[2 + perm[i]];
            Gp[i] = (const float*)d_in[14 + perm[i]];
            Wp[i] = (const float*)d_in[26 + perm[i]];
        }
    }

    const int* nbr1 = (const int*)d_in[38];
    const int* tbl2 = (const int*)d_in[39];
    const int* nbr2 = (const int*)d_in[40];
    const int* tbl3 = (const int*)d_in[41];
    const int* nbr3 = (const int*)d_in[42];
    const int* tbl4 = (const int*)d_in[43];
    const int* nbr4 = (const int*)d_in[44];
    const int* tbl_out = (const int*)d_in[45];
    const int* pix = (const int*)d_in[46];

    const int N1 = in_sizes[0] / 4;
    const int N2 = in_sizes[39] / 27;
    const int N3 = in_sizes[41] / 27;
    const int N4 = in_sizes[43] / 27;
    const int NO = in_sizes[45] / 3;

    float* ws = (float*)d_ws;
    float* xA = ws;                      // (N1,16)
    float* xB = xA + (size_t)N1 * 16;    // (N1,16)
    float* x2a = xB + (size_t)N1 * 16;   // (N2,32)
    float* x2b = x2a + (size_t)N2 * 32;  // (N2,32)
    float* x3a = x2b + (size_t)N2 * 32;  // (N3,64)
    float* x3b = x3a + (size_t)N3 * 64;  // (N3,64)
    float* x4a = x3b + (size_t)N3 * 64;  // (N4,64)
    float* x4b = x4a + (size_t)N4 * 64;  // (N4,64)
    float* outp = (float*)d_out;         // (NO,128)

    // Packed-weight regions after the feature buffers.
    static const int dims[12][3] = {
        {27, 4, 16},  {27, 16, 16}, {27, 16, 32}, {27, 32, 32},
        {27, 32, 32}, {27, 32, 64}, {27, 64, 64}, {27, 64, 64},
        {27, 64, 64}, {27, 64, 64}, {27, 64, 64}, {3, 64, 128}};
    float* wpk[12];
    float* wcur = x4b + (size_t)N4 * 64;
    for (int i = 0; i < 12; ++i) {
        wpk[i] = wcur;
        wcur += (size_t)dims[i][0] * dims[i][1] * dims[i][2];
    }
    for (int i = 0; i < 12; ++i) {
        int total = dims[i][0] * dims[i][1] * dims[i][2];
        repack_w<<<(total + 255) / 256, 256, 0, stream>>>(
            Wp[i], wpk[i], dims[i][0], dims[i][1], dims[i][2]);
    }

    LAUNCH_SPCONV(4, 16, 27, voxel_feats, nbr1, wpk[0], Gp[0], Bp[0], xA, N1);  // in
    LAUNCH_SPCONV(16, 16, 27, xA, nbr1, wpk[1], Gp[1], Bp[1], xB, N1);          // c1
    {
        int threads = N1 * 16;
        img_fuse<<<(threads + 255) / 256, 256, 0, stream>>>(xB, img, pix, N1);
    }
    LAUNCH_SPCONV(16, 32, 27, xB, tbl2, wpk[2], Gp[2], Bp[2], x2a, N2);          // c2a
    LAUNCH_SPCONV(32, 32, 27, x2a, nbr2, wpk[3], Gp[3], Bp[3], x2b, N2);         // c2b
    LAUNCH_SPCONV(32, 32, 27, x2b, nbr2, wpk[4], Gp[4], Bp[4], x2a, N2);         // c2c
    LAUNCH_SPCONV(32, 64, 27, x2a, tbl3, wpk[5], Gp[5], Bp[5], x3a, N3);         // c3a
    LAUNCH_SPCONV(64, 64, 27, x3a, nbr3, wpk[6], Gp[6], Bp[6], x3b, N3);         // c3b
    LAUNCH_SPCONV(64, 64, 27, x3b, nbr3, wpk[7], Gp[7], Bp[7], x3a, N3);         // c3c
    LAUNCH_SPCONV(64, 64, 27, x3a, tbl4, wpk[8], Gp[8], Bp[8], x4a, N4);         // c4a
    LAUNCH_SPCONV(64, 64, 27, x4a, nbr4, wpk[9], Gp[9], Bp[9], x4b, N4);         // c4b
    LAUNCH_SPCONV(64, 64, 27, x4b, nbr4, wpk[10], Gp[10], Bp[10], x4a, N4);      // c4c
    LAUNCH_SPCONV(64, 128, 3, x4a, tbl_out, wpk[11], Gp[11], Bp[11], outp, NO);  // out
}